// FragNetLayer_58033598103709
// MI455X (gfx1250) — compile-verified
//
#include <hip/hip_runtime.h>
#include <hip/hip_bf16.h>

typedef __attribute__((ext_vector_type(2))) float v2f;
typedef __attribute__((ext_vector_type(8))) float v8f;

#define D128 128
#define CONDK 16

// ---------------------------------------------------------------------------
// Generic fill
// ---------------------------------------------------------------------------
__global__ void fill_kernel(float* __restrict__ p, float v, long n) {
    long i = (long)blockIdx.x * blockDim.x + threadIdx.x;
    if (i < n) p[i] = v;
}

// ---------------------------------------------------------------------------
// Degree: deg starts at 1.0 (self loop), count src occurrences
// ---------------------------------------------------------------------------
__global__ void deg_count_kernel(const int* __restrict__ src, float* __restrict__ deg, int E) {
    int e = blockIdx.x * blockDim.x + threadIdx.x;
    if (e < E) atomicAdd(&deg[src[e]], 1.0f);
}

__global__ void rsqrt_inplace_kernel(float* __restrict__ deg, int n) {
    int i = blockIdx.x * blockDim.x + threadIdx.x;
    if (i < n) {
        float d = deg[i];
        deg[i] = (d > 0.0f) ? rsqrtf(d) : 0.0f;
    }
}

// ---------------------------------------------------------------------------
// FP32 WMMA GEMM:  C[M,Nc] = act(A[M,K] @ B[K,Nc] + bias)
// One wave computes one 16x16 output tile via V_WMMA_F32_16X16X4_F32.
// Block = 256 threads = 8 waves -> 8 adjacent column tiles (128 cols).
// All branches that affect WMMA are wave-uniform so EXEC stays all-ones.
//
// VGPR layouts (CDNA5 ISA 7.12.2):
//  A 16x4 f32 : lane L holds row M=L%16, cols K = kb, kb+1 where kb=(L/16)*2
//  B 4x16 f32 : lane L holds col N=L%16, rows K = kb, kb+1
//  C 16x16 f32: vgpr r, lanes 0-15 -> M=r, lanes 16-31 -> M=r+8, N=L%16
// ---------------------------------------------------------------------------
__global__ __launch_bounds__(256)
void wmma_gemm_f32_kernel(const float* __restrict__ A, const float* __restrict__ B,
                          const float* __restrict__ bias, float* __restrict__ C,
                          int M, int K, int Nc, int relu) {
    const int wave = threadIdx.x >> 5;
    const int lane = threadIdx.x & 31;
    const int m0 = blockIdx.x * 16;
    const int n0 = (blockIdx.y * (blockDim.x >> 5) + wave) * 16;
    if (n0 >= Nc || m0 >= M) return;                 // wave-uniform exit

    const int row    = m0 + (lane & 15);
    const int rclamp = (row < M) ? row : (M - 1);    // clamped load, guarded store
    const int kb     = (lane >> 4) << 1;             // 0 or 2
    const int ncol   = n0 + (lane & 15);

    v8f c = {};
    const float* ap = A + (size_t)rclamp * K + kb;
    for (int k = 0; k < K; k += 4) {
        v2f a, b;
        a[0] = ap[k];
        a[1] = ap[k + 1];
        const float* bp = B + (size_t)(k + kb) * Nc + ncol;
        b[0] = bp[0];
        b[1] = bp[Nc];
        c = __builtin_amdgcn_wmma_f32_16x16x4_f32(false, a, false, b,
                                                  (short)0, c, false, false);
    }

    const float bv = bias ? bias[ncol] : 0.0f;
#pragma unroll
    for (int r = 0; r < 8; ++r) {
        int mrow = m0 + r + ((lane >> 4) << 3);
        if (mrow < M) {
            float v = c[r] + bv;
            if (relu) v = fmaxf(v, 0.0f);
            C[(size_t)mrow * Nc + ncol] = v;
        }
    }
}

// ---------------------------------------------------------------------------
// Edge message scatter: out[tgt] += x[src] * (dis[src]*dis[tgt])
// One wave per edge; each lane moves a float4 (32 lanes * 16B = 512B row).
// ---------------------------------------------------------------------------
__global__ __launch_bounds__(256)
void edge_scatter_kernel(const int* __restrict__ src, const int* __restrict__ tgt,
                         const float* __restrict__ dis, const float* __restrict__ x,
                         float* __restrict__ out, int E) {
    int e = blockIdx.x * (blockDim.x >> 5) + (threadIdx.x >> 5);
    if (e >= E) return;
    int lane = threadIdx.x & 31;
    int s = src[e];
    int t = tgt[e];
    float norm = dis[s] * dis[t];
    const float4 v = ((const float4*)(x + (size_t)s * D128))[lane];
    float* op = out + (size_t)t * D128 + lane * 4;
    atomicAdd(op + 0, v.x * norm);
    atomicAdd(op + 1, v.y * norm);
    atomicAdd(op + 2, v.z * norm);
    atomicAdd(op + 3, v.w * norm);
}

// ---------------------------------------------------------------------------
// Fragment edge scatter (unnormalized): fsum[ft] += xf[fs]
// ---------------------------------------------------------------------------
__global__ __launch_bounds__(256)
void frag_scatter_kernel(const int* __restrict__ fs, const int* __restrict__ ft,
                         const float* __restrict__ xf, float* __restrict__ fsum, int EF) {
    int e = blockIdx.x * (blockDim.x >> 5) + (threadIdx.x >> 5);
    if (e >= EF) return;
    int lane = threadIdx.x & 31;
    int s = fs[e];
    int t = ft[e];
    const float4 v = ((const float4*)(xf + (size_t)s * D128))[lane];
    float* op = fsum + (size_t)t * D128 + lane * 4;
    atomicAdd(op + 0, v.x);
    atomicAdd(op + 1, v.y);
    atomicAdd(op + 2, v.z);
    atomicAdd(op + 3, v.w);
}

// ---------------------------------------------------------------------------
// FiLM (K=16 affine from cond) with optional fused self-loop term and
// optional fused atom->fragment pooling.
//   acc  = val[i,d] (+ selfx[i,d]*dis[i]^2 if selfx)
//   out  = (bg[d] + cond[i,:]@Wg[:,d]) * acc + (bb[d] + cond[i,:]@Wb[:,d])
//   pool[a2f[i], d] += out   (if pool)
// ---------------------------------------------------------------------------
__global__ __launch_bounds__(256)
void film_kernel(const float* __restrict__ val, const float* __restrict__ selfx,
                 const float* __restrict__ dis, const float* __restrict__ cond,
                 const float* __restrict__ Wg, const float* __restrict__ bg,
                 const float* __restrict__ Wb, const float* __restrict__ bb,
                 const int* __restrict__ a2f, float* __restrict__ out,
                 float* __restrict__ pool, int rows) {
    long tid = (long)blockIdx.x * blockDim.x + threadIdx.x;
    if (tid >= (long)rows * D128) return;
    int i = (int)(tid >> 7);
    int d = (int)(tid & 127);

    float acc = val[tid];
    if (selfx) {
        float ds = dis[i];
        acc = fmaf(selfx[tid], ds * ds, acc);
    }
    float g = bg[d];
    float b = bb[d];
    const float* cr = cond + (size_t)i * CONDK;
#pragma unroll
    for (int c = 0; c < CONDK; ++c) {
        float cv = cr[c];
        g = fmaf(cv, Wg[c * D128 + d], g);
        b = fmaf(cv, Wb[c * D128 + d], b);
    }
    float r = fmaf(g, acc, b);
    out[tid] = r;
    if (pool) atomicAdd(&pool[(size_t)a2f[i] * D128 + d], r);
}

// ---------------------------------------------------------------------------
// Launcher
// ---------------------------------------------------------------------------
static inline int ceil_div(long a, long b) { return (int)((a + b - 1) / b); }

extern "C" void kernel_launch(void* const* d_in, const int* in_sizes, int n_in,
                              void* d_out, int out_size, void* d_ws, size_t ws_size,
                              hipStream_t stream) {
    const float* x_atoms   = (const float*)d_in[0];
    const int*   edge_idx  = (const int*)  d_in[1];
    /* d_in[2] edge_attr: unused by reference forward */
    const int*   frag_idx  = (const int*)  d_in[3];
    /* d_in[4] x_frags: only used for shape */
    const int*   a2f       = (const int*)  d_in[5];
    const float* cond_atom = (const float*)d_in[6];
    const float* cond_frag = (const float*)d_in[7];
    const float* W_atom    = (const float*)d_in[8];
    const float* b_atom    = (const float*)d_in[9];
    const float* W_f1      = (const float*)d_in[10];
    const float* b_f1      = (const float*)d_in[11];
    const float* W_f2      = (const float*)d_in[12];
    const float* b_f2      = (const float*)d_in[13];
    const float* Wg_a      = (const float*)d_in[14];
    const float* bg_a      = (const float*)d_in[15];
    const float* Wb_a      = (const float*)d_in[16];
    const float* bb_a      = (const float*)d_in[17];
    const float* Wg_f      = (const float*)d_in[18];
    const float* bg_f      = (const float*)d_in[19];
    const float* Wb_f      = (const float*)d_in[20];
    const float* bb_f      = (const float*)d_in[21];

    const int N  = in_sizes[0] / D128;
    const int E  = in_sizes[1] / 2;
    const int F  = in_sizes[4] / D128;
    const int EF = in_sizes[3] / 2;

    const int* src = edge_idx;
    const int* tgt = edge_idx + E;
    const int* fs  = frag_idx;
    const int* ft  = frag_idx + EF;

    float* out_atoms = (float*)d_out;                  // [N,128]
    float* out_frags = out_atoms + (size_t)N * D128;   // [F,128]

    float* ws       = (float*)d_ws;
    float* x_embed  = ws;  ws += (size_t)N * D128;     // [N,128]
    float* deg      = ws;  ws += (size_t)N;            // [N] -> becomes deg_inv_sqrt
    float* xf       = ws;  ws += (size_t)F * D128;     // [F,128]
    float* frag_sum = ws;  ws += (size_t)F * D128;     // [F,128]
    float* h1       = ws;  ws += (size_t)F * 2 * D128; // [F,256]

    const int BT = 256;

    // 1) init accumulators
    fill_kernel<<<ceil_div((long)N * D128, BT), BT, 0, stream>>>(out_atoms, 0.0f, (long)N * D128);
    fill_kernel<<<ceil_div((long)F * D128, BT), BT, 0, stream>>>(xf, 0.0f, (long)F * D128);
    fill_kernel<<<ceil_div((long)F * D128, BT), BT, 0, stream>>>(frag_sum, 0.0f, (long)F * D128);
    fill_kernel<<<ceil_div(N, BT), BT, 0, stream>>>(deg, 1.0f, N);   // self loop contributes 1

    // 2) degree + inv-sqrt
    deg_count_kernel<<<ceil_div(E, BT), BT, 0, stream>>>(src, deg, E);
    rsqrt_inplace_kernel<<<ceil_div(N, BT), BT, 0, stream>>>(deg, N);

    // 3) atom embed GEMM: x_embed = x_atoms @ W_atom + b_atom   (WMMA f32 16x16x4)
    {
        dim3 grid(ceil_div(N, 16), 1);
        wmma_gemm_f32_kernel<<<grid, BT, 0, stream>>>(x_atoms, W_atom, b_atom, x_embed,
                                                      N, D128, D128, 0);
    }

    // 4) normalized edge scatter: out_atoms[tgt] += x_embed[src] * dis[src]*dis[tgt]
    edge_scatter_kernel<<<ceil_div(E, BT / 32), BT, 0, stream>>>(src, tgt, deg, x_embed,
                                                                 out_atoms, E);

    // 5) FiLM on atoms (+ fused self-loop term) + fused atom->frag pooling
    film_kernel<<<ceil_div((long)N * D128, BT), BT, 0, stream>>>(
        out_atoms, x_embed, deg, cond_atom, Wg_a, bg_a, Wb_a, bb_a, a2f,
        out_atoms, xf, N);

    // 6) fragment message passing: frag_sum[ft] += xf[fs]
    frag_scatter_kernel<<<ceil_div(EF, BT / 32), BT, 0, stream>>>(fs, ft, xf, frag_sum, EF);

    // 7) frag MLP layer 1 (fused bias+ReLU): h1 = relu(frag_sum @ W_f1 + b_f1)
    {
        dim3 grid(ceil_div(F, 16), 2);   // Nc = 256 -> two column groups of 128
        wmma_gemm_f32_kernel<<<grid, BT, 0, stream>>>(frag_sum, W_f1, b_f1, h1,
                                                      F, D128, 2 * D128, 1);
    }

    // 8) frag MLP layer 2 -> straight into d_out frag region
    {
        dim3 grid(ceil_div(F, 16), 1);
        wmma_gemm_f32_kernel<<<grid, BT, 0, stream>>>(h1, W_f2, b_f2, out_frags,
                                                      F, 2 * D128, D128, 0);
    }

    // 9) FiLM on fragments, in place on d_out
    film_kernel<<<ceil_div((long)F * D128, BT), BT, 0, stream>>>(
        out_frags, nullptr, nullptr, cond_frag, Wg_f, bg_f, Wb_f, bb_f, nullptr,
        out_frags, nullptr, F);
}